// BasicGCN_39273180955610
// MI455X (gfx1250) — compile-verified
//
#include <hip/hip_runtime.h>
#include <math.h>

#define N_NODES 50000
#define N_EDGES 800000
#define F_DIM   128
#define H_DIM   128
#define C_DIM   10
#define G_DIM   64
#define N_ROWTILES (N_NODES / 16)   // 3125, exact

typedef __attribute__((ext_vector_type(2))) float v2f;
typedef __attribute__((ext_vector_type(8))) float v8f;

// ---------------------------------------------------------------------------
// Dense GEMM  out[N,H] = A[N,F] @ W[F,H]  using V_WMMA_F32_16X16X4_F32.
// One wave -> one 16-row block x all 8 column tiles (16 cols each).
// A lane layout (16x4 f32): lane L holds row M=L%16, K = 2*(L/16) + {0,1}.
// B lane layout (4x16 f32): lane L holds col N=L%16, K = 2*(L/16) + {0,1}.
// C/D (16x16 f32, 8 VGPRs): VGPR r = rows {r, r+8}, col = lane%16.
// ---------------------------------------------------------------------------
__global__ __launch_bounds__(128)
void gemm_f32_wmma(const float* __restrict__ A, const float* __restrict__ W,
                   float* __restrict__ out, int nRowTiles) {
    const int wave = threadIdx.x >> 5;
    const int lane = threadIdx.x & 31;
    const int rowTile = blockIdx.x * 4 + wave;
    if (rowTile >= nRowTiles) return;          // whole-wave guard: EXEC stays all-1s

    const int m     = lane & 15;               // row (A) / col (B,C,D) within tile
    const int khalf = (lane >> 4) << 1;        // 0 or 2
    const float* __restrict__ Arow = A + (size_t)(rowTile * 16 + m) * F_DIM;

    v8f acc[8];
#pragma unroll
    for (int t = 0; t < 8; ++t)
#pragma unroll
        for (int r = 0; r < 8; ++r) acc[t][r] = 0.0f;

    for (int k = 0; k < F_DIM; k += 4) {
        const int kk = k + khalf;
        v2f a;
        a.x = Arow[kk];
        a.y = Arow[kk + 1];
#pragma unroll
        for (int t = 0; t < 8; ++t) {
            const int col = t * 16 + m;
            v2f b;
            b.x = W[(size_t)kk * H_DIM + col];
            b.y = W[(size_t)(kk + 1) * H_DIM + col];
            acc[t] = __builtin_amdgcn_wmma_f32_16x16x4_f32(
                false, a, false, b, (short)0, acc[t], false, false);
        }
    }

    const int rowBase = rowTile * 16 + ((lane >> 4) << 3);  // +8 for upper half-lanes
#pragma unroll
    for (int t = 0; t < 8; ++t)
#pragma unroll
        for (int r = 0; r < 8; ++r)
            out[(size_t)(rowBase + r) * H_DIM + t * 16 + m] = acc[t][r];
}

// ---------------------------------------------------------------------------
__global__ void zero_k(float* p, int n) {
    int i = blockIdx.x * blockDim.x + threadIdx.x;
    if (i < n) p[i] = 0.0f;
}

__global__ void deg_k(const int* __restrict__ dst, float* __restrict__ deg) {
    int e = blockIdx.x * blockDim.x + threadIdx.x;
    if (e < N_EDGES) atomicAdd(&deg[dst[e]], 1.0f);
}

__global__ void isd_k(float* d) {   // deg -> 1/sqrt(deg+1), in place
    int i = blockIdx.x * blockDim.x + threadIdx.x;
    if (i < N_NODES) d[i] = rsqrtf(d[i] + 1.0f);
}

__global__ void coef_k(const int* __restrict__ src, const int* __restrict__ dst,
                       const float* __restrict__ isd, float* __restrict__ coef) {
    int e = blockIdx.x * blockDim.x + threadIdx.x;
    if (e < N_EDGES) coef[e] = isd[src[e]] * isd[dst[e]];
}

// agg = isd^2 * hw + bias (self-loop + bias = accumulator init), float4-wide
__global__ void init_agg_k(const float4* __restrict__ hw, const float* __restrict__ isd,
                           const float4* __restrict__ b, float4* __restrict__ agg) {
    int idx = blockIdx.x * blockDim.x + threadIdx.x;   // over N * 32 float4s
    if (idx >= N_NODES * (H_DIM / 4)) return;
    int i = idx >> 5, q = idx & 31;
    float s = isd[i];
    float s2 = s * s;
    float4 h = hw[idx];
    float4 bb = b[q];
    float4 r;
    r.x = s2 * h.x + bb.x;  r.y = s2 * h.y + bb.y;
    r.z = s2 * h.z + bb.z;  r.w = s2 * h.w + bb.w;
    agg[idx] = r;
}

// agg[dst] += coef[e] * hw[src] -- one thread per (edge, 4 features); a wave's
// 32 lanes cover exactly one edge, so src/dst/coef loads are wave-broadcast.
__global__ void scatter_k(const int* __restrict__ src, const int* __restrict__ dst,
                          const float* __restrict__ coef, const float* __restrict__ hw,
                          float* __restrict__ agg) {
    int idx = blockIdx.x * blockDim.x + threadIdx.x;   // over E * 32
    if (idx >= N_EDGES * (H_DIM / 4)) return;
    int e = idx >> 5, q = (idx & 31) << 2;
    float c = coef[e];
    const float4 h = *(const float4*)(hw + (size_t)src[e] * H_DIM + q);
    float* ap = agg + (size_t)dst[e] * H_DIM + q;
    atomicAdd(ap + 0, c * h.x);
    atomicAdd(ap + 1, c * h.y);
    atomicAdd(ap + 2, c * h.z);
    atomicAdd(ap + 3, c * h.w);
}

__global__ void relu_k(float4* p, int n4) {
    int i = blockIdx.x * blockDim.x + threadIdx.x;
    if (i >= n4) return;
    float4 v = p[i];
    v.x = fmaxf(v.x, 0.0f);  v.y = fmaxf(v.y, 0.0f);
    v.z = fmaxf(v.z, 0.0f);  v.w = fmaxf(v.w, 0.0f);
    p[i] = v;
}

__global__ void counts_k(const int* __restrict__ batch, float* __restrict__ counts) {
    int i = blockIdx.x * blockDim.x + threadIdx.x;
    if (i < N_NODES) atomicAdd(&counts[batch[i]], 1.0f);
}

// per-graph sum + max; values >= 0 post-ReLU so integer atomicMax on the float
// bit pattern is exact (0-init matches the reference's where(counts>0,...,0)).
__global__ void pool_scatter_k(const float* __restrict__ h, const int* __restrict__ batch,
                               float* __restrict__ sums, int* __restrict__ mx) {
    int idx = blockIdx.x * blockDim.x + threadIdx.x;   // over N * 32
    if (idx >= N_NODES * (H_DIM / 4)) return;
    int i = idx >> 5, q = (idx & 31) << 2;
    int g = batch[i];
    const float4 v = *(const float4*)(h + (size_t)i * H_DIM + q);
    float* sp = sums + g * H_DIM + q;
    int*   mp = mx   + g * H_DIM + q;
    atomicAdd(sp + 0, v.x);  atomicMax(mp + 0, __float_as_int(v.x));
    atomicAdd(sp + 1, v.y);  atomicMax(mp + 1, __float_as_int(v.y));
    atomicAdd(sp + 2, v.z);  atomicMax(mp + 2, __float_as_int(v.z));
    atomicAdd(sp + 3, v.w);  atomicMax(mp + 3, __float_as_int(v.w));
}

__global__ void pool_finalize_k(const float* __restrict__ sums, const int* __restrict__ mx,
                                const float* __restrict__ counts, float* __restrict__ aggr) {
    int idx = blockIdx.x * blockDim.x + threadIdx.x;
    if (idx >= G_DIM * 2 * H_DIM) return;
    int g = idx / (2 * H_DIM);
    int j = idx % (2 * H_DIM);
    float v = (j < H_DIM) ? sums[g * H_DIM + j] / fmaxf(counts[g], 1.0f)
                          : __int_as_float(mx[g * H_DIM + (j - H_DIM)]);
    aggr[idx] = v;
}

// out[G,C] = aggr[G,2H] @ Wa[2H,C] + ba   (tiny: 640 dot products of length 256)
__global__ void out_k(const float* __restrict__ aggr, const float* __restrict__ Wa,
                      const float* __restrict__ ba, float* __restrict__ out) {
    int idx = blockIdx.x * blockDim.x + threadIdx.x;
    if (idx >= G_DIM * C_DIM) return;
    int g = idx / C_DIM, c = idx % C_DIM;
    float s = ba[c];
    for (int j = 0; j < 2 * H_DIM; ++j)
        s += aggr[g * 2 * H_DIM + j] * Wa[j * C_DIM + c];
    out[idx] = s;
}

// ---------------------------------------------------------------------------
extern "C" void kernel_launch(void* const* d_in, const int* in_sizes, int n_in,
                              void* d_out, int out_size, void* d_ws, size_t ws_size,
                              hipStream_t stream) {
    const float* x     = (const float*)d_in[0];
    const int*   ei    = (const int*)  d_in[1];
    const int*   batch = (const int*)  d_in[2];
    const float* W0 = (const float*)d_in[3];  const float* b0 = (const float*)d_in[4];
    const float* W1 = (const float*)d_in[5];  const float* b1 = (const float*)d_in[6];
    const float* W2 = (const float*)d_in[7];  const float* b2 = (const float*)d_in[8];
    const float* Wa = (const float*)d_in[9];  const float* ba = (const float*)d_in[10];

    const int* src = ei;
    const int* dst = ei + N_EDGES;

    float* out  = (float*)d_out;              // [G, C] first
    float* aggr = out + G_DIM * C_DIM;        // then [G, 2H]

    // workspace carve-out (256B aligned slices)
    char* ws = (char*)d_ws;
    auto carve = [&](size_t bytes) -> void* {
        void* p = (void*)ws;
        ws += (bytes + 255) & ~(size_t)255;
        return p;
    };
    float* isd    = (float*)carve((size_t)N_NODES * 4);
    float* coef   = (float*)carve((size_t)N_EDGES * 4);
    float* hw     = (float*)carve((size_t)N_NODES * H_DIM * 4);
    float* agg    = (float*)carve((size_t)N_NODES * H_DIM * 4);
    float* sums   = (float*)carve((size_t)G_DIM * H_DIM * 4);
    int*   mx     = (int*)  carve((size_t)G_DIM * H_DIM * 4);
    float* counts = (float*)carve((size_t)G_DIM * 4);

    const int NB   = 256;
    const int nNH4 = N_NODES * (H_DIM / 4);   // 1.6M float4s
    const int nEH4 = N_EDGES * (H_DIM / 4);   // 25.6M threads

    // degrees -> inv sqrt deg -> edge coefficients (layer-invariant)
    zero_k<<<(N_NODES + NB - 1) / NB, NB, 0, stream>>>(isd, N_NODES);
    deg_k <<<(N_EDGES + NB - 1) / NB, NB, 0, stream>>>(dst, isd);
    isd_k <<<(N_NODES + NB - 1) / NB, NB, 0, stream>>>(isd);
    coef_k<<<(N_EDGES + NB - 1) / NB, NB, 0, stream>>>(src, dst, isd, coef);

    const float* Ws[3] = {W0, W1, W2};
    const float* bs[3] = {b0, b1, b2};
    const float* in = x;
    for (int l = 0; l < 3; ++l) {
        gemm_f32_wmma<<<(N_ROWTILES + 3) / 4, 128, 0, stream>>>(in, Ws[l], hw, N_ROWTILES);
        init_agg_k<<<(nNH4 + NB - 1) / NB, NB, 0, stream>>>(
            (const float4*)hw, isd, (const float4*)bs[l], (float4*)agg);
        scatter_k<<<(nEH4 + NB - 1) / NB, NB, 0, stream>>>(src, dst, coef, hw, agg);
        relu_k<<<(nNH4 + NB - 1) / NB, NB, 0, stream>>>((float4*)agg, nNH4);
        in = agg;
    }

    // pooling
    zero_k<<<(G_DIM * H_DIM + NB - 1) / NB, NB, 0, stream>>>(sums, G_DIM * H_DIM);
    zero_k<<<(G_DIM * H_DIM + NB - 1) / NB, NB, 0, stream>>>((float*)mx, G_DIM * H_DIM);
    zero_k<<<1, NB, 0, stream>>>(counts, G_DIM);
    counts_k<<<(N_NODES + NB - 1) / NB, NB, 0, stream>>>(batch, counts);
    pool_scatter_k<<<(nNH4 + NB - 1) / NB, NB, 0, stream>>>(agg, batch, sums, mx);
    pool_finalize_k<<<(G_DIM * 2 * H_DIM + NB - 1) / NB, NB, 0, stream>>>(sums, mx, counts, aggr);

    // classifier
    out_k<<<(G_DIM * C_DIM + NB - 1) / NB, NB, 0, stream>>>(aggr, Wa, ba, out);
}